// Threshold_weights4_52699248721947
// MI455X (gfx1250) — compile-verified
//
#include <hip/hip_runtime.h>
#include <math.h>

typedef float v4f __attribute__((ext_vector_type(4)));

#define WAVES_PER_BLOCK 8
#define BLOCK_THREADS (WAVES_PER_BLOCK * 32)

// One wave32 per row. Streams all 5 logit matrices for that row with
// non-temporal b128 loads (5 concurrent HBM streams per wave), keeps a
// per-lane running top-2 via v_med3_f32/v_max_f32, then does a wave32
// butterfly top-2 merge and the 5-way softmax. The target-class logit is
// fetched with one broadcast load per matrix (no per-element compare).
// Also produces a per-block partial max over outputs1..4 for max_preds.
__global__ __launch_bounds__(BLOCK_THREADS)
void margin_softmax_kernel(const float* __restrict__ o1,
                           const float* __restrict__ o2,
                           const float* __restrict__ o3,
                           const float* __restrict__ o4,
                           const float* __restrict__ mi,
                           const int*   __restrict__ targets,
                           float* __restrict__ out,       // [1 + N*5]
                           float* __restrict__ blockmax,  // [gridDim.x]
                           int N, int C) {
  const int lane = threadIdx.x & 31;
  const int wv   = threadIdx.x >> 5;
  const int row  = blockIdx.x * WAVES_PER_BLOCK + wv;
  __shared__ float smax[WAVES_PER_BLOCK];

  float rowmax4 = -INFINITY;  // max over outputs1..4 for this row

  if (row < N) {
    const size_t roff = (size_t)row * (size_t)C;
    const float* base[5] = { o1 + roff, o2 + roff, o3 + roff, o4 + roff, mi + roff };
    const int tcol = targets[row];

    // Target-class logit: one broadcast 4B load per matrix, issued first
    // so it's in flight while the streaming loop runs. Wave-uniform.
    float tg[5];
#pragma unroll
    for (int m = 0; m < 5; ++m) tg[m] = base[m][tcol];

    float m1[5], m2[5];
#pragma unroll
    for (int m = 0; m < 5; ++m) { m1[m] = -INFINITY; m2[m] = -INFINITY; }

    const int nf4 = C >> 2;                       // C = 1000 -> 250 float4 per row
    for (int f4 = lane; f4 < nf4; f4 += 32) {     // coalesced: 32 lanes x 16B = 512B
#pragma unroll
      for (int m = 0; m < 5; ++m) {
        v4f v = __builtin_nontemporal_load((const v4f*)(base[m]) + f4);
#pragma unroll
        for (int j = 0; j < 4; ++j) {
          float x = v[j];
          // top-2 update: m2 = med3(x, m1, m2); m1 = max(m1, x)
          m2[m] = __builtin_amdgcn_fmed3f(x, m1[m], m2[m]);
          m1[m] = fmaxf(m1[m], x);
        }
      }
    }
    // scalar tail (C % 4 != 0); no-op for C = 1000
    for (int c = (nf4 << 2) + lane; c < C; c += 32) {
#pragma unroll
      for (int m = 0; m < 5; ++m) {
        float x = __builtin_nontemporal_load(base[m] + c);
        m2[m] = __builtin_amdgcn_fmed3f(x, m1[m], m2[m]);
        m1[m] = fmaxf(m1[m], x);
      }
    }

    // wave32 butterfly top-2 merge
#pragma unroll
    for (int off = 16; off > 0; off >>= 1) {
#pragma unroll
      for (int m = 0; m < 5; ++m) {
        float a1 = __shfl_xor(m1[m], off, 32);
        float a2 = __shfl_xor(m2[m], off, 32);
        float hi = fmaxf(m1[m], a1);
        float lo = fminf(m1[m], a1);
        m1[m] = hi;
        m2[m] = fmaxf(lo, fmaxf(m2[m], a2));
      }
    }

    // margin = top1 - top2 iff target attains the max, else 0
    float margin[5];
#pragma unroll
    for (int m = 0; m < 5; ++m)
      margin[m] = (tg[m] == m1[m]) ? (m1[m] - m2[m]) : 0.0f;

    // softmax(margin / T), T = 2
    float mx = margin[0] * 0.5f;
#pragma unroll
    for (int m = 1; m < 5; ++m) mx = fmaxf(mx, margin[m] * 0.5f);
    float e[5], sum = 0.0f;
#pragma unroll
    for (int m = 0; m < 5; ++m) { e[m] = __expf(margin[m] * 0.5f - mx); sum += e[m]; }
    const float inv = 1.0f / sum;
    if (lane < 5) out[1 + (size_t)row * 5 + lane] = e[lane] * inv;

    rowmax4 = fmaxf(fmaxf(m1[0], m1[1]), fmaxf(m1[2], m1[3]));
  }

  // block-level partial max over outputs1..4 (mimic excluded)
  if (lane == 0) smax[wv] = rowmax4;  // -inf for inactive waves
  __syncthreads();
  if (threadIdx.x == 0) {
    float v = smax[0];
#pragma unroll
    for (int w = 1; w < WAVES_PER_BLOCK; ++w) v = fmaxf(v, smax[w]);
    blockmax[blockIdx.x] = v;
  }
}

// Single-block finisher: reduce per-block partial maxes -> d_out[0].
// Deterministic (no atomics), independent of d_out's prior contents.
__global__ __launch_bounds__(256)
void final_max_kernel(const float* __restrict__ part, int n, float* __restrict__ out) {
  __shared__ float sm[256];
  float v = -INFINITY;
  for (int i = threadIdx.x; i < n; i += 256) v = fmaxf(v, part[i]);
  sm[threadIdx.x] = v;
  __syncthreads();
  for (int s = 128; s > 0; s >>= 1) {
    if (threadIdx.x < s) sm[threadIdx.x] = fmaxf(sm[threadIdx.x], sm[threadIdx.x + s]);
    __syncthreads();
  }
  if (threadIdx.x == 0) out[0] = sm[0];
}

extern "C" void kernel_launch(void* const* d_in, const int* in_sizes, int n_in,
                              void* d_out, int out_size, void* d_ws, size_t ws_size,
                              hipStream_t stream) {
  const float* o1 = (const float*)d_in[0];
  const float* o2 = (const float*)d_in[1];
  const float* o3 = (const float*)d_in[2];
  const float* o4 = (const float*)d_in[3];
  const float* mi = (const float*)d_in[4];
  const int* targets = (const int*)d_in[5];

  const int N = in_sizes[5];            // 16384
  const int C = in_sizes[0] / N;        // 1000

  float* out  = (float*)d_out;          // [0]=max_preds, [1..N*5]=out_threshold
  float* bmax = (float*)d_ws;           // gridDim partial maxes

  const int grid = (N + WAVES_PER_BLOCK - 1) / WAVES_PER_BLOCK;

  margin_softmax_kernel<<<grid, BLOCK_THREADS, 0, stream>>>(
      o1, o2, o3, o4, mi, targets, out, bmax, N, C);
  final_max_kernel<<<1, 256, 0, stream>>>(bmax, grid, out);
}